// MultiHopResGATLayer_40785009443442
// MI455X (gfx1250) — compile-verified
//
#include <hip/hip_runtime.h>
#include <hip/hip_bf16.h>

typedef float v2f __attribute__((ext_vector_type(2)));
typedef float v8f __attribute__((ext_vector_type(8)));

#define NN 50000
#define EE 800000
#define LRELU_SLOPE 0.2f
#define LN_EPS 1e-5f

__device__ __forceinline__ float lrelu(float x) { return x > 0.0f ? x : LRELU_SLOPE * x; }

// Butterfly lane exchange via ds_swizzle_b32 (group-of-32 XOR mode):
// offset = xor_mask << 10 | and_mask(0x1f).  Single DS op, no index setup.
template <int XOR_MASK>
__device__ __forceinline__ float swz_xor_add(float v) {
    int i = __builtin_amdgcn_ds_swizzle(__float_as_int(v), (XOR_MASK << 10) | 0x1f);
    return v + __int_as_float(i);
}

// ---------------------------------------------------------------------------
// Zero int buffer
// ---------------------------------------------------------------------------
__global__ void zero_kernel(int* __restrict__ p, int n) {
    int i = blockIdx.x * blockDim.x + threadIdx.x;
    if (i < n) p[i] = 0;
}

// ---------------------------------------------------------------------------
// GEMM + fused attention logits.
// xw = x @ W via V_WMMA_F32_16X16X4_F32; grid (3125, 2), block (32, 8).
// Each block: one 16-row strip of x staged in LDS; wave w = head w = columns
// [16w, 16w+16); blockIdx.y selects the W0/W1 projection.
// A (16x4 f32): lanes0-15 M=lane K={k,k+1}; lanes16-31 K={k+2,k+3}.
// B (4x16 f32): VGPR0 lanes0-15 row k, lanes16-31 row k+2; VGPR1 = +1.
// D (16x16 f32): VGPR v -> M = v + 8*(lane>=16), N = lane&15.
// Epilogue: a_s[n,h] = <d[n, h*16..], att_src[h,..]> reduced across the 16
// lanes of each half-wave with ds_swizzle XOR steps 1,2,4,8 (stay in-half).
// ---------------------------------------------------------------------------
__global__ __launch_bounds__(256)
void gemm_wmma_kernel(const float* __restrict__ x,
                      const float* __restrict__ W0,
                      const float* __restrict__ W1,
                      const float* __restrict__ as0, const float* __restrict__ ad0,
                      const float* __restrict__ as1, const float* __restrict__ ad1,
                      float* __restrict__ xw0, float* __restrict__ xw1,
                      float* __restrict__ as0n, float* __restrict__ ad0n,
                      float* __restrict__ as1n, float* __restrict__ ad1n) {
    __shared__ float lx[16 * 260];              // padded stride: no bank conflicts
    const int row0 = blockIdx.x * 16;
    const float* __restrict__ W   = blockIdx.y ? W1 : W0;
    const float* __restrict__ ATS = blockIdx.y ? as1 : as0;
    const float* __restrict__ ATD = blockIdx.y ? ad1 : ad0;
    float* __restrict__ XW  = blockIdx.y ? xw1 : xw0;
    float* __restrict__ ASN = blockIdx.y ? as1n : as0n;
    float* __restrict__ ADN = blockIdx.y ? ad1n : ad0n;

    const int t = threadIdx.y * 32 + threadIdx.x;
    for (int i = t; i < 16 * 256; i += 256) {
        int r = i >> 8, c = i & 255;
        lx[r * 260 + c] = x[(row0 + r) * 256 + c];
    }
    __syncthreads();

    const int lane = threadIdx.x;
    const int n0   = threadIdx.y * 16;          // column tile == head*16
    const int mrow = lane & 15;
    const int koff = (lane >> 4) * 2;

    v8f c = {};
    for (int k = 0; k < 256; k += 4) {
        v2f a, b;
        a.x = lx[mrow * 260 + k + koff];
        a.y = lx[mrow * 260 + k + koff + 1];
        b.x = W[(k + koff) * 128 + n0 + mrow];
        b.y = W[(k + koff + 1) * 128 + n0 + mrow];
        c = __builtin_amdgcn_wmma_f32_16x16x4_f32(false, a, false, b,
                                                  (short)0, c, false, false);
    }

    const int colN  = n0 + mrow;
    const int rbase = row0 + ((lane >> 4) << 3);
    const float asv = ATS[n0 + mrow];           // att_src[head][channel]
    const float adv = ATD[n0 + mrow];
    const int head  = threadIdx.y;
#pragma unroll
    for (int v = 0; v < 8; ++v) {
        XW[(rbase + v) * 128 + colN] = c[v];
        float ps = c[v] * asv;
        float pd = c[v] * adv;
        ps = swz_xor_add<1>(ps);  pd = swz_xor_add<1>(pd);
        ps = swz_xor_add<2>(ps);  pd = swz_xor_add<2>(pd);
        ps = swz_xor_add<4>(ps);  pd = swz_xor_add<4>(pd);
        ps = swz_xor_add<8>(ps);  pd = swz_xor_add<8>(pd);
        if (mrow == 0) {
            ASN[(rbase + v) * 8 + head] = ps;
            ADN[(rbase + v) * 8 + head] = pd;
        }
    }
}

// ---------------------------------------------------------------------------
// Degree histogram for hop1 (dst = col[i]) and hop2 (dst = col[col[i]]).
// ---------------------------------------------------------------------------
__global__ void hist_kernel(const int* __restrict__ col,
                            int* __restrict__ cnt0, int* __restrict__ cnt1) {
    int i = blockIdx.x * blockDim.x + threadIdx.x;
    if (i >= EE) return;
    int j = col[i];                 // hop1 dst
    atomicAdd(&cnt0[j], 1);
    atomicAdd(&cnt1[col[j]], 1);    // hop2 dst
}

// ---------------------------------------------------------------------------
// Single-block exclusive scan of both count arrays -> CSR offsets.
// Zeroes counts for reuse as scatter cursors.
// ---------------------------------------------------------------------------
__global__ __launch_bounds__(1024)
void scan_kernel(int* __restrict__ cnt0, int* __restrict__ off0,
                 int* __restrict__ cnt1, int* __restrict__ off1, int n) {
    __shared__ int tmp[1024];
    const int tid = threadIdx.x;
    for (int arr = 0; arr < 2; ++arr) {
        int* cnt = arr ? cnt1 : cnt0;
        int* off = arr ? off1 : off0;
        int carry = 0;
        for (int base = 0; base < n; base += 1024) {
            int i = base + tid;
            int v = (i < n) ? cnt[i] : 0;
            tmp[tid] = v;
            __syncthreads();
            for (int o = 1; o < 1024; o <<= 1) {
                int t2 = (tid >= o) ? tmp[tid - o] : 0;
                __syncthreads();
                tmp[tid] += t2;
                __syncthreads();
            }
            if (i < n) { off[i] = carry + tmp[tid] - v; cnt[i] = 0; }
            int tot = tmp[1023];
            __syncthreads();
            carry += tot;
        }
        if (tid == 0) off[n] = carry;
        __syncthreads();
    }
}

// ---------------------------------------------------------------------------
// Scatter src node ids into CSR slots (cursor = zeroed counts).
// ---------------------------------------------------------------------------
__global__ void scatter_kernel(const int* __restrict__ row, const int* __restrict__ col,
                               const int* __restrict__ off0, int* __restrict__ cur0,
                               int* __restrict__ src0,
                               const int* __restrict__ off1, int* __restrict__ cur1,
                               int* __restrict__ src1) {
    int i = blockIdx.x * blockDim.x + threadIdx.x;
    if (i >= EE) return;
    int j = col[i];
    int p0 = off0[j] + atomicAdd(&cur0[j], 1);
    src0[p0] = row[i];
    int d2 = col[j];
    int p1 = off1[d2] + atomicAdd(&cur1[d2], 1);
    src1[p1] = row[j];
}

// ---------------------------------------------------------------------------
// Fused per-node softmax aggregation (both hops) + bias + residual + LayerNorm.
// One 64-thread block per node: wave 0 = hop0, wave 1 = hop1.
// Within a wave: 32 lanes = 8 heads x 4 lanes; each lane owns a float4 of the
// 128 hop channels -> thread t = hop*32+lane owns output channels [4t, 4t+4).
// Self loop handled analytically (never stored in CSR).  LN statistics are
// reduced with ds_swizzle butterflies + a 4-word LDS cross-wave exchange.
// ---------------------------------------------------------------------------
__global__ __launch_bounds__(64)
void aggregate_ln_kernel(const float* __restrict__ xw0, const float* __restrict__ xw1,
                         const float* __restrict__ as0n, const float* __restrict__ ad0n,
                         const float* __restrict__ as1n, const float* __restrict__ ad1n,
                         const int* __restrict__ off0, const int* __restrict__ src0,
                         const int* __restrict__ off1, const int* __restrict__ src1,
                         const float* __restrict__ b0, const float* __restrict__ b1,
                         const float* __restrict__ x,
                         const float* __restrict__ gamma, const float* __restrict__ beta,
                         float* __restrict__ out) {
    __shared__ float red[4];                    // {sum, sumsq} per wave
    const int node = blockIdx.x;
    const int lane = threadIdx.x;               // 0..31
    const int hop  = threadIdx.y;               // 0..1

    const float* __restrict__ XW   = hop ? xw1 : xw0;
    const float* __restrict__ AS   = hop ? as1n : as0n;
    const float* __restrict__ AD   = hop ? ad1n : ad0n;
    const int*   __restrict__ off  = hop ? off1 : off0;
    const int*   __restrict__ src  = hop ? src1 : src0;
    const float* __restrict__ bias = hop ? b1 : b0;

    const int head = lane >> 2;
    const float ad = AD[node * 8 + head];
    const float e_self = lrelu(AS[node * 8 + head] + ad);
    const int beg = off[node], end = off[node + 1];

    // pass 1: segment max (incl. self loop)
    float m = e_self;
    for (int i = beg; i < end; ++i) {
        int s = src[i];
        m = fmaxf(m, lrelu(AS[s * 8 + head] + ad));
    }

    // pass 2: exp-sum + weighted accumulate
    const float4* __restrict__ XWv = (const float4*)XW;
    float p = __expf(e_self - m);
    float ssum = p;
    float4 w = XWv[node * 32 + lane];
    float ax = p * w.x, ay = p * w.y, az = p * w.z, aw = p * w.w;
    for (int i = beg; i < end; ++i) {
        int s = src[i];
        float pe = __expf(lrelu(AS[s * 8 + head] + ad) - m);
        ssum += pe;
        float4 v = XWv[s * 32 + lane];
        ax += pe * v.x; ay += pe * v.y; az += pe * v.z; aw += pe * v.w;
    }
    const float inv = 1.0f / ssum;
    const float4 bv = ((const float4*)bias)[lane];

    // this thread's 4 output channels: h + bias + residual
    const int t = hop * 32 + lane;              // channel group 0..63
    const float4 xv = ((const float4*)(x + node * 256))[t];
    float4 v;
    v.x = ax * inv + bv.x + xv.x;
    v.y = ay * inv + bv.y + xv.y;
    v.z = az * inv + bv.z + xv.z;
    v.w = aw * inv + bv.w + xv.w;

    // LayerNorm statistics: ds_swizzle butterfly within wave, LDS across waves
    float s  = v.x + v.y + v.z + v.w;
    float sq = v.x * v.x + v.y * v.y + v.z * v.z + v.w * v.w;
    s = swz_xor_add<1>(s);   sq = swz_xor_add<1>(sq);
    s = swz_xor_add<2>(s);   sq = swz_xor_add<2>(sq);
    s = swz_xor_add<4>(s);   sq = swz_xor_add<4>(sq);
    s = swz_xor_add<8>(s);   sq = swz_xor_add<8>(sq);
    s = swz_xor_add<16>(s);  sq = swz_xor_add<16>(sq);
    if (lane == 0) { red[hop * 2] = s; red[hop * 2 + 1] = sq; }
    __syncthreads();
    const float S    = red[0] + red[2];
    const float SQ   = red[1] + red[3];
    const float mu   = S * (1.0f / 256.0f);
    const float var  = SQ * (1.0f / 256.0f) - mu * mu;
    const float rstd = rsqrtf(var + LN_EPS);

    const float4 gg = ((const float4*)gamma)[t];
    const float4 bb = ((const float4*)beta)[t];
    float4 r;
    r.x = (v.x - mu) * rstd * gg.x + bb.x;
    r.y = (v.y - mu) * rstd * gg.y + bb.y;
    r.z = (v.z - mu) * rstd * gg.z + bb.z;
    r.w = (v.w - mu) * rstd * gg.w + bb.w;
    ((float4*)(out + node * 256))[t] = r;
}

// ---------------------------------------------------------------------------
extern "C" void kernel_launch(void* const* d_in, const int* in_sizes, int n_in,
                              void* d_out, int out_size, void* d_ws, size_t ws_size,
                              hipStream_t stream) {
    const float* x   = (const float*)d_in[0];
    const int*   ei  = (const int*)d_in[1];      // [2,E] int32
    const float* W0  = (const float*)d_in[2];
    const float* as0 = (const float*)d_in[3];
    const float* ad0 = (const float*)d_in[4];
    const float* b0  = (const float*)d_in[5];
    const float* W1  = (const float*)d_in[6];
    const float* as1 = (const float*)d_in[7];
    const float* ad1 = (const float*)d_in[8];
    const float* b1  = (const float*)d_in[9];
    const float* gamma = (const float*)d_in[10];
    const float* beta  = (const float*)d_in[11];
    float* out = (float*)d_out;

    const int* row = ei;
    const int* col = ei + EE;

    // workspace layout
    float* xw0  = (float*)d_ws;             // N*128
    float* xw1  = xw0  + NN * 128;          // N*128
    float* as0n = xw1  + NN * 128;          // N*8
    float* ad0n = as0n + NN * 8;
    float* as1n = ad0n + NN * 8;
    float* ad1n = as1n + NN * 8;
    int* cnt0 = (int*)(ad1n + NN * 8);      // N
    int* cnt1 = cnt0 + NN;                  // N
    int* off0 = cnt1 + NN;                  // N+1
    int* off1 = off0 + (NN + 1);            // N+1
    int* src0 = off1 + (NN + 1);            // E
    int* src1 = src0 + EE;                  // E

    // 1. zero counters
    zero_kernel<<<(2 * NN + 255) / 256, 256, 0, stream>>>(cnt0, 2 * NN);

    // 2. dense projections (WMMA) + fused attention logits
    gemm_wmma_kernel<<<dim3(NN / 16, 2), dim3(32, 8), 0, stream>>>(
        x, W0, W1, as0, ad0, as1, ad1, xw0, xw1, as0n, ad0n, as1n, ad1n);

    // 3-5. CSR build for both hops
    hist_kernel<<<(EE + 255) / 256, 256, 0, stream>>>(col, cnt0, cnt1);
    scan_kernel<<<1, 1024, 0, stream>>>(cnt0, off0, cnt1, off1, NN);
    scatter_kernel<<<(EE + 255) / 256, 256, 0, stream>>>(row, col, off0, cnt0, src0,
                                                         off1, cnt1, src1);

    // 6. fused softmax aggregation + bias + residual + LayerNorm
    aggregate_ln_kernel<<<dim3(NN, 1), dim3(32, 2), 0, stream>>>(
        xw0, xw1, as0n, ad0n, as1n, ad1n,
        off0, src0, off1, src1, b0, b1, x, gamma, beta, out);
}